// AttendModel_shared_80564996538677
// MI455X (gfx1250) — compile-verified
//
#include <hip/hip_runtime.h>

typedef float v2f __attribute__((ext_vector_type(2)));
typedef float v8f __attribute__((ext_vector_type(8)));

#define ESTRIDE 130   // LDS row pitch in floats (pad 128 -> 130 to avoid bank conflicts)

__device__ __forceinline__ float half16_sum(float t) {
  // xor masks < 16 never cross the 16-lane half of a wave32
  t += __shfl_xor(t, 1);
  t += __shfl_xor(t, 2);
  t += __shfl_xor(t, 4);
  t += __shfl_xor(t, 8);
  return t;
}

// ---------------------------------------------------------------------------
// Kernel A: per 8-batch-row tile -> lane embeds, sr ref, sq GEMM (WMMA fp32),
// attention logits a[b, l] written to workspace.
// block = 128 threads (4 waves). smem = 79360 bytes.
// ---------------------------------------------------------------------------
__global__ __launch_bounds__(128) void attn_logits_kernel(
    const float* __restrict__ states,
    const float* __restrict__ le_w, const float* __restrict__ le_b,
    const float* __restrict__ sq_w, const float* __restrict__ sq_b,
    const float* __restrict__ sr_w, const float* __restrict__ sr_b,
    const float* __restrict__ sa_w, const float* __restrict__ sa_b,
    float* __restrict__ lgws) {
  extern __shared__ char smem[];
  float* E  = (float*)(smem);                    // [128][130]
  float* xs = (float*)(smem + 66560);            // [128][8]
  float* mm = (float*)(smem + 66560 + 4096);     // [8][128]
  float* rr = (float*)(smem + 66560 + 8192);     // [8][128]
  float* lg = (float*)(smem + 66560 + 12288);    // [128]

  const int tid = threadIdx.x;
  const int b0  = blockIdx.x * 8;

  // stage batch tile: states rows are 129 floats (phase + 16*8 demands)
  for (int i = tid; i < 8 * 129; i += 128) {
    const int bs = i / 129, j = i % 129;
    const float v = states[(b0 + bs) * 129 + j];
    if (j > 0) xs[bs * 128 + (j - 1)] = v;
  }
  lg[tid] = 0.0f;
  __syncthreads();

  // lane embeds E[m][c] (c = tid), and per-batch mean
  float lw[8];
#pragma unroll
  for (int d = 0; d < 8; ++d) lw[d] = le_w[tid * 8 + d];
  const float lb = le_b[tid];
  for (int bs = 0; bs < 8; ++bs) {
    float s = 0.0f;
#pragma unroll
    for (int l = 0; l < 16; ++l) {
      const int m = bs * 16 + l;
      float e = lb;
#pragma unroll
      for (int d = 0; d < 8; ++d) e += xs[m * 8 + d] * lw[d];
      E[m * ESTRIDE + tid] = e;
      s += e;
    }
    mm[bs * 128 + tid] = s * 0.0625f;
  }
  __syncthreads();

  // lane_ref rr[bs][c] = mean @ sr_w^T + sr_b
  {
    float acc[8];
#pragma unroll
    for (int bs = 0; bs < 8; ++bs) acc[bs] = sr_b[tid];
    for (int k = 0; k < 128; ++k) {
      const float wv = sr_w[tid * 128 + k];
#pragma unroll
      for (int bs = 0; bs < 8; ++bs) acc[bs] += mm[bs * 128 + k] * wv;
    }
#pragma unroll
    for (int bs = 0; bs < 8; ++bs) rr[bs * 128 + tid] = acc[bs];
  }
  __syncthreads();

  // sq GEMM: M=128 (8 rows x 16 lanes), N=128, K=128 on V_WMMA_F32_16X16X4_F32
  const int wid = tid >> 5, lid = tid & 31, hf = lid >> 4, lc = lid & 15;
  const float sab = sa_b[0];
  for (int ntb = 0; ntb < 2; ++ntb) {
    const int n = (wid * 2 + ntb) * 16 + lc;
    v2f Bt[32];
    const float* wp = sq_w + n * 128 + 2 * hf;   // B[k][n] = sq_w[n][k]
#pragma unroll
    for (int j = 0; j < 32; ++j) Bt[j] = *(const v2f*)(wp + 4 * j);
    const float sbn = sq_b[n];
    const float saw = sa_w[n];
    for (int mt = 0; mt < 8; ++mt) {
      v8f c = {};
      const float* ep = E + (mt * 16 + lc) * ESTRIDE + 2 * hf;
#pragma unroll
      for (int j = 0; j < 32; ++j) {
        const v2f a = *(const v2f*)(ep + 4 * j);
        c = __builtin_amdgcn_wmma_f32_16x16x4_f32(false, a, false, Bt[j],
                                                  (short)0, c, false, false);
      }
      const float rv = rr[mt * 128 + n] + sbn;   // ref + sq bias (per-column)
#pragma unroll
      for (int v = 0; v < 8; ++v) {
        float t = tanhf(c[v] + rv) * saw;        // sa_w dot, partial over 16 n's
        t = half16_sum(t);
        if (lc == 0) atomicAdd(&lg[mt * 16 + v + 8 * hf], t);
      }
    }
  }
  __syncthreads();

  {
    const int bs = tid >> 4, l = tid & 15;
    lgws[(b0 + bs) * 16 + l] = lg[tid] + sab;
  }
}

// ---------------------------------------------------------------------------
// Kernel B: softmax-over-batch stats: per lane l -> max_b a[b,l], sum_b exp(.)
// ---------------------------------------------------------------------------
__global__ __launch_bounds__(1024) void softmax_stats_kernel(
    const float* __restrict__ lgws, float* __restrict__ mxsum, int B) {
  __shared__ float s[1024];
  const int t = threadIdx.x, l = t & 15, g = t >> 4;  // 64 groups x 16 lanes
  float mx = -3.4e38f;
  for (int r = g; r < B; r += 64) mx = fmaxf(mx, lgws[r * 16 + l]);
  s[t] = mx;
  __syncthreads();
  for (int st = 512; st >= 16; st >>= 1) {
    if (t < st) s[t] = fmaxf(s[t], s[t + st]);
    __syncthreads();
  }
  const float m = s[l];
  __syncthreads();
  float sm = 0.0f;
  for (int r = g; r < B; r += 64) sm += expf(lgws[r * 16 + l] - m);
  s[t] = sm;
  __syncthreads();
  for (int st = 512; st >= 16; st >>= 1) {
    if (t < st) s[t] += s[t + st];
    __syncthreads();
  }
  if (t < 16) { mxsum[t] = m; mxsum[16 + t] = s[t]; }
}

// ---------------------------------------------------------------------------
// Kernel C: attn weights -> pairs, phase gather, ar (VALU), aq GEMM (WMMA),
// tanh + aa dot -> out[B,8]. block = 128 threads. smem = 108832 bytes.
// ---------------------------------------------------------------------------
__global__ __launch_bounds__(128) void attend_out_kernel(
    const float* __restrict__ states,
    const float* __restrict__ le_w, const float* __restrict__ le_b,
    const float* __restrict__ aq_w, const float* __restrict__ aq_b,
    const float* __restrict__ ar_w, const float* __restrict__ ar_b,
    const float* __restrict__ aa_w, const float* __restrict__ aa_b,
    const int* __restrict__ pp,
    const float* __restrict__ lgws, const float* __restrict__ mxsum,
    float* __restrict__ out) {
  extern __shared__ char smem[];
  float* E      = (float*)(smem);                      // [128][130]
  float* P      = (float*)(smem + 66560);              // [64][130] pairs
  float* xs     = (float*)(smem + 66560 + 33280);      // [128][8]
  float* gg     = (float*)(smem + 66560 + 37376);      // [8][128] green@ar
  float* attn_s = (float*)(smem + 66560 + 41472);      // [8][16]
  float* qacc   = (float*)(smem + 66560 + 41984);      // [64]
  int*   ph_s   = (int*)  (smem + 66560 + 42240);      // [8]

  const int tid = threadIdx.x;
  const int b0  = blockIdx.x * 8;

  for (int i = tid; i < 8 * 129; i += 128) {
    const int bs = i / 129, j = i % 129;
    const float v = states[(b0 + bs) * 129 + j];
    if (j == 0) ph_s[bs] = (int)v;
    else        xs[bs * 128 + (j - 1)] = v;
  }
  {  // attn = exp(a - max)/sum, per (bs, lane)
    const int bs = tid >> 4, l = tid & 15;
    const float a = lgws[(b0 + bs) * 16 + l];
    attn_s[tid] = expf(a - mxsum[l]) / mxsum[16 + l];
  }
  if (tid < 64) qacc[tid] = 0.0f;
  __syncthreads();

  // recompute lane embeds (cheap: avoids 34MB workspace round-trip)
  {
    float lw[8];
#pragma unroll
    for (int d = 0; d < 8; ++d) lw[d] = le_w[tid * 8 + d];
    const float lb = le_b[tid];
    for (int bs = 0; bs < 8; ++bs) {
#pragma unroll
      for (int l = 0; l < 16; ++l) {
        const int m = bs * 16 + l;
        float e = lb;
#pragma unroll
        for (int d = 0; d < 8; ++d) e += xs[m * 8 + d] * lw[d];
        E[m * ESTRIDE + tid] = e;
      }
    }
  }
  __syncthreads();

  // pairs P[bs*8+p][c] = attn[i0]*E[i0] + attn[i1]*E[i1]
  {
    int p0[8], p1[8];
#pragma unroll
    for (int p = 0; p < 8; ++p) { p0[p] = pp[2 * p]; p1[p] = pp[2 * p + 1]; }
    for (int bs = 0; bs < 8; ++bs) {
#pragma unroll
      for (int p = 0; p < 8; ++p) {
        const int i0 = p0[p], i1 = p1[p];
        P[(bs * 8 + p) * ESTRIDE + tid] =
            attn_s[bs * 16 + i0] * E[(bs * 16 + i0) * ESTRIDE + tid] +
            attn_s[bs * 16 + i1] * E[(bs * 16 + i1) * ESTRIDE + tid];
      }
    }
  }
  __syncthreads();

  // green = P[bs][phase] @ ar_w^T + ar_b
  {
    int ph[8];
#pragma unroll
    for (int bs = 0; bs < 8; ++bs) ph[bs] = ph_s[bs];
    float acc[8];
#pragma unroll
    for (int bs = 0; bs < 8; ++bs) acc[bs] = ar_b[tid];
    for (int k = 0; k < 128; ++k) {
      const float wv = ar_w[tid * 128 + k];
#pragma unroll
      for (int bs = 0; bs < 8; ++bs)
        acc[bs] += P[(bs * 8 + ph[bs]) * ESTRIDE + k] * wv;
    }
#pragma unroll
    for (int bs = 0; bs < 8; ++bs) gg[bs * 128 + tid] = acc[bs];
  }
  __syncthreads();

  // aq GEMM: M=64, N=128, K=128 on WMMA fp32; tanh + aa dot folded
  const int wid = tid >> 5, lid = tid & 31, hf = lid >> 4, lc = lid & 15;
  for (int ntb = 0; ntb < 2; ++ntb) {
    const int n = (wid * 2 + ntb) * 16 + lc;
    v2f Bt[32];
    const float* wp = aq_w + n * 128 + 2 * hf;
#pragma unroll
    for (int j = 0; j < 32; ++j) Bt[j] = *(const v2f*)(wp + 4 * j);
    const float aqn = aq_b[n];
    const float aaw = aa_w[n];
    for (int mt = 0; mt < 4; ++mt) {
      v8f c = {};
      const float* ap = P + (mt * 16 + lc) * ESTRIDE + 2 * hf;
#pragma unroll
      for (int j = 0; j < 32; ++j) {
        const v2f a = *(const v2f*)(ap + 4 * j);
        c = __builtin_amdgcn_wmma_f32_16x16x4_f32(false, a, false, Bt[j],
                                                  (short)0, c, false, false);
      }
      const int bs = mt * 2 + hf;                 // uniform per lane in tile
      const float gv = gg[bs * 128 + n] + aqn;
#pragma unroll
      for (int v = 0; v < 8; ++v) {
        float t = tanhf(c[v] + gv) * aaw;
        t = half16_sum(t);
        if (lc == 0) atomicAdd(&qacc[mt * 16 + v + 8 * hf], t);
      }
    }
  }
  __syncthreads();

  if (tid < 64)
    out[(b0 + (tid >> 3)) * 8 + (tid & 7)] = qacc[tid] + aa_b[0];
}

// ---------------------------------------------------------------------------
extern "C" void kernel_launch(void* const* d_in, const int* in_sizes, int n_in,
                              void* d_out, int out_size, void* d_ws, size_t ws_size,
                              hipStream_t stream) {
  const float* states = (const float*)d_in[0];
  const float* le_w = (const float*)d_in[1];
  const float* le_b = (const float*)d_in[2];
  const float* sq_w = (const float*)d_in[3];
  const float* sq_b = (const float*)d_in[4];
  const float* sr_w = (const float*)d_in[5];
  const float* sr_b = (const float*)d_in[6];
  const float* sa_w = (const float*)d_in[7];
  const float* sa_b = (const float*)d_in[8];
  const float* aq_w = (const float*)d_in[9];
  const float* aq_b = (const float*)d_in[10];
  const float* ar_w = (const float*)d_in[11];
  const float* ar_b = (const float*)d_in[12];
  const float* aa_w = (const float*)d_in[13];
  const float* aa_b = (const float*)d_in[14];
  const int*   pp   = (const int*)d_in[15];
  float* outp = (float*)d_out;

  const int B = in_sizes[0] / 129;          // 16384
  const int nblk = B / 8;

  float* lgws  = (float*)d_ws;              // [B*16] logits
  float* mxsum = lgws + (size_t)B * 16;     // [16] max, [16] sum

  const size_t smemA = 66560 + 4096 + 4096 + 4096 + 512;           // 79360
  const size_t smemC = 66560 + 33280 + 4096 + 4096 + 512 + 256 + 32; // 108832

  attn_logits_kernel<<<nblk, 128, smemA, stream>>>(
      states, le_w, le_b, sq_w, sq_b, sr_w, sr_b, sa_w, sa_b, lgws);
  softmax_stats_kernel<<<1, 1024, 0, stream>>>(lgws, mxsum, B);
  attend_out_kernel<<<nblk, 128, smemC, stream>>>(
      states, le_w, le_b, aq_w, aq_b, ar_w, ar_b, aa_w, aa_b, pp,
      lgws, mxsum, outp);
}